// ReinforceAgentQuantum_29875792511858
// MI455X (gfx1250) — compile-verified
//
#include <hip/hip_runtime.h>
#include <hip/hip_bf16.h>

#define NQ   12
#define DIM  4096
#define BATCH 2048
#define NACT 4

typedef __attribute__((ext_vector_type(2))) float v2f;
typedef __attribute__((ext_vector_type(8))) float v8f;

__device__ __forceinline__ v8f wmma4(v2f a, v2f b, v8f c) {
    // D = A(16x4) * B(4x16) + C ; f32 everywhere
    return __builtin_amdgcn_wmma_f32_16x16x4_f32(false, a, false, b, (short)0, c, false, false);
}

// State-index mapping for the three 4-wire groups (tile t, row r, col c)
// GROUP 0: wires 0-3  (high nibble)  -> state(16,256): idx = r*256 + t*16 + c
// GROUP 1: wires 4-7  (mid nibble)   -> state(16,16,16): idx = t*256 + r*16 + c
// GROUP 2: wires 8-11 (low nibble)   -> state(256,16): idx = (t*16+r)*16 + c
template<int GROUP>
__device__ __forceinline__ int sidx(int t, int r, int c) {
    if (GROUP == 0) return r * 256 + t * 16 + c;
    if (GROUP == 1) return t * 256 + r * 16 + c;
    return (t * 16 + r) * 16 + c;
}

// Apply 16x16 (complex or real) unitary U to the state via WMMA.
// GROUP 0/1: D = U @ S_tile ; GROUP 2: D = S_tile @ U^T
// 16 independent tiles; 8 waves x 2 tiles each. All reads into VGPRs complete
// before writeback, and tiles are disjoint, so no intra-group barrier needed.
template<int GROUP, bool REALU>
__device__ void apply_group(float* sRe, float* sIm, const float* uRe, const float* uIm) {
    const int lane = threadIdx.x & 31;
    const int wave = threadIdx.x >> 5;
    const int n  = lane & 15;   // A-matrix M index and B/D N index
    const int hi = lane >> 4;   // K-chunk half selector

    for (int tt = 0; tt < 2; ++tt) {
        const int t = wave * 2 + tt;
        v8f dre = {0.f,0.f,0.f,0.f,0.f,0.f,0.f,0.f};
        v8f dim_ = {0.f,0.f,0.f,0.f,0.f,0.f,0.f,0.f};
#pragma unroll
        for (int kc = 0; kc < 4; ++kc) {
            const int k0 = kc * 4 + 2 * hi;
            const int k1 = k0 + 1;
            if (GROUP != 2) {
                // A = U (row n), B = S tile (col n)
                v2f aur; aur.x = uRe[n * 16 + k0]; aur.y = uRe[n * 16 + k1];
                v2f bsr; bsr.x = sRe[sidx<GROUP>(t, k0, n)]; bsr.y = sRe[sidx<GROUP>(t, k1, n)];
                v2f bsi; bsi.x = sIm[sidx<GROUP>(t, k0, n)]; bsi.y = sIm[sidx<GROUP>(t, k1, n)];
                dre  = wmma4(aur, bsr, dre);    // Ur*Sr
                dim_ = wmma4(aur, bsi, dim_);   // Ur*Si
                if (!REALU) {
                    v2f aui; aui.x = uIm[n * 16 + k0]; aui.y = uIm[n * 16 + k1];
                    v2f nai = -aui;
                    dre  = wmma4(nai, bsi, dre);   // -Ui*Si
                    dim_ = wmma4(aui, bsr, dim_);  //  Ui*Sr
                }
            } else {
                // A = S tile (row n), B = U^T (col n): B(K,N)=U[n][K]
                v2f asr; asr.x = sRe[sidx<2>(t, n, k0)]; asr.y = sRe[sidx<2>(t, n, k1)];
                v2f asi; asi.x = sIm[sidx<2>(t, n, k0)]; asi.y = sIm[sidx<2>(t, n, k1)];
                v2f bur; bur.x = uRe[n * 16 + k0]; bur.y = uRe[n * 16 + k1];
                dre  = wmma4(asr, bur, dre);    // Sr*Ur^T
                dim_ = wmma4(asi, bur, dim_);   // Si*Ur^T
                if (!REALU) {
                    v2f bui; bui.x = uIm[n * 16 + k0]; bui.y = uIm[n * 16 + k1];
                    v2f nbi = -bui;
                    dre  = wmma4(asi, nbi, dre);   // -Si*Ui^T
                    dim_ = wmma4(asr, bui, dim_);  //  Sr*Ui^T
                }
            }
        }
#pragma unroll
        for (int v = 0; v < 8; ++v) {
            const int M = v + 8 * hi;
            sRe[sidx<GROUP>(t, M, n)] = dre[v];
            sIm[sidx<GROUP>(t, M, n)] = dim_[v];
        }
    }
}

// Fused 4-wire RX Kronecker unitary: U[r][c] = mag * (-i)^popcount(r^c)
__device__ void build_u_rx(float* uRe, float* uIm, const float* x, const float* iscal,
                           int b, int layer, int g) {
    const int tid = threadIdx.x;
    const int r = tid >> 4, c = tid & 15;
    float mag = 1.f;
#pragma unroll
    for (int q = 0; q < 4; ++q) {
        const int w = g * 4 + q;
        const float th = 0.5f * iscal[layer * NQ + w] * x[b * NQ + w];
        const int p = 3 - q;
        mag *= (((r >> p) & 1) == ((c >> p) & 1)) ? __cosf(th) : __sinf(th);
    }
    float re, im;
    switch (__popc(r ^ c) & 3) {
        case 0:  re =  mag; im = 0.f;  break;
        case 1:  re = 0.f;  im = -mag; break;
        case 2:  re = -mag; im = 0.f;  break;
        default: re = 0.f;  im =  mag; break;
    }
    uRe[tid] = re; uIm[tid] = im;
}

// Fused 4-wire RY Kronecker unitary (real): sign = (-1)^{#(r=0,c=1) bits}
__device__ void build_u_ry(float* uRe, const float* wts, int layer, int g) {
    const int tid = threadIdx.x;
    const int r = tid >> 4, c = tid & 15;
    float val = 1.f;
#pragma unroll
    for (int q = 0; q < 4; ++q) {
        const int w = g * 4 + q;
        const float th = 0.5f * wts[layer * 24 + 12 + w];
        const int p = 3 - q;
        const int rb = (r >> p) & 1, cb = (c >> p) & 1;
        if (rb == cb) val *= __cosf(th);
        else          val *= (rb ? __sinf(th) : -__sinf(th));
    }
    uRe[tid] = val;
}

// JAX threefry2x32, key = jax.random.key(42) -> (0, 42)
__device__ void threefry2x32(unsigned x0, unsigned x1, unsigned& o0, unsigned& o1) {
    const unsigned ks0 = 0u, ks1 = 42u;
    const unsigned ks2 = 0x1BD11BDAu ^ ks0 ^ ks1;
    const unsigned ks[3] = {ks0, ks1, ks2};
    const int R[8] = {13, 15, 26, 6, 17, 29, 16, 24};
    x0 += ks0; x1 += ks1;
#pragma unroll
    for (int g = 0; g < 5; ++g) {
#pragma unroll
        for (int i = 0; i < 4; ++i) {
            const int rot = R[(g & 1) * 4 + i];
            x0 += x1;
            x1 = (x1 << rot) | (x1 >> (32 - rot));
            x1 ^= x0;
        }
        x0 += ks[(g + 1) % 3];
        x1 += ks[(g + 2) % 3] + (unsigned)(g + 1);
    }
    o0 = x0; o1 = x1;
}

__global__ __launch_bounds__(256)
void pqc_reinforce_kernel(const float* __restrict__ x, const float* __restrict__ iscal,
                          const float* __restrict__ wts, const float* __restrict__ oscal,
                          float* __restrict__ out) {
    const int b = blockIdx.x;
    const int tid = threadIdx.x;

    __shared__ float sRe[DIM];
    __shared__ float sIm[DIM];
    __shared__ float uRe[256];
    __shared__ float uIm[256];
    __shared__ float eacc[NACT];

    // |0...0>
#pragma unroll
    for (int j = 0; j < 16; ++j) {
        const int idx = tid + 256 * j;
        sRe[idx] = (idx == 0) ? 1.f : 0.f;
        sIm[idx] = 0.f;
    }
    if (tid < NACT) eacc[tid] = 0.f;
    __syncthreads();

    for (int layer = 0; layer < 2; ++layer) {
        // ---- 12 RX gates as three 16x16 complex WMMA unitaries ----
        build_u_rx(uRe, uIm, x, iscal, b, layer, 0); __syncthreads();
        apply_group<0, false>(sRe, sIm, uRe, uIm);   __syncthreads();
        build_u_rx(uRe, uIm, x, iscal, b, layer, 1); __syncthreads();
        apply_group<1, false>(sRe, sIm, uRe, uIm);   __syncthreads();
        build_u_rx(uRe, uIm, x, iscal, b, layer, 2); __syncthreads();
        apply_group<2, false>(sRe, sIm, uRe, uIm);   __syncthreads();

        // ---- 12 RZ gates fused into one diagonal phase pass ----
#pragma unroll
        for (int j = 0; j < 16; ++j) {
            const int idx = tid + 256 * j;
            float ang = 0.f;
#pragma unroll
            for (int w = 0; w < NQ; ++w)
                ang += wts[layer * 24 + w] * ((((idx >> (11 - w)) & 1) != 0) ? 0.5f : -0.5f);
            const float cp = __cosf(ang), sp = __sinf(ang);
            const float re = sRe[idx], im = sIm[idx];
            sRe[idx] = re * cp - im * sp;
            sIm[idx] = re * sp + im * cp;
        }
        __syncthreads();

        // ---- 12 RY gates as three 16x16 real WMMA unitaries ----
        build_u_ry(uRe, wts, layer, 0); __syncthreads();
        apply_group<0, true>(sRe, sIm, uRe, uIm); __syncthreads();
        build_u_ry(uRe, wts, layer, 1); __syncthreads();
        apply_group<1, true>(sRe, sIm, uRe, uIm); __syncthreads();
        build_u_ry(uRe, wts, layer, 2); __syncthreads();
        apply_group<2, true>(sRe, sIm, uRe, uIm); __syncthreads();

        // ---- CZ ring: sign = (-1)^{# adjacent (ring) bit pairs both 1} ----
#pragma unroll
        for (int j = 0; j < 16; ++j) {
            const int idx = tid + 256 * j;
            const unsigned rot = ((unsigned)(idx << 1) | ((unsigned)idx >> 11)) & 0xFFFu;
            if (__popc((unsigned)idx & rot) & 1) {
                sRe[idx] = -sRe[idx];
                sIm[idx] = -sIm[idx];
            }
        }
        __syncthreads();
    }

    // ---- PauliZ expvals on wires 0..3 ----
    float e[NACT] = {0.f, 0.f, 0.f, 0.f};
#pragma unroll
    for (int j = 0; j < 16; ++j) {
        const int idx = tid + 256 * j;
        const float re = sRe[idx], im = sIm[idx];
        const float p = re * re + im * im;
#pragma unroll
        for (int w = 0; w < NACT; ++w)
            e[w] += (((idx >> (11 - w)) & 1) != 0) ? -p : p;
    }
#pragma unroll
    for (int off = 16; off > 0; off >>= 1)
#pragma unroll
        for (int w = 0; w < NACT; ++w)
            e[w] += __shfl_down(e[w], off, 32);
    if ((tid & 31) == 0)
        for (int w = 0; w < NACT; ++w) atomicAdd(&eacc[w], e[w]);
    __syncthreads();

    // ---- categorical sample (Gumbel-argmax, threefry) + log_softmax ----
    if (tid == 0) {
        float logits[NACT];
        for (int jj = 0; jj < NACT; ++jj) logits[jj] = eacc[jj] * oscal[jj];

        float best = -3.4e38f; int a = 0;
        for (int jj = 0; jj < NACT; ++jj) {
            const unsigned i = (unsigned)(b * NACT + jj);  // flat counter, shape (B, NACT)
            unsigned x0, x1; int word;
            const unsigned half = (unsigned)(BATCH * NACT / 2);
            if (i < half) { x0 = i; x1 = i + half; word = 0; }
            else          { x0 = i - half; x1 = i; word = 1; }
            unsigned y0, y1;
            threefry2x32(x0, x1, y0, y1);
            const unsigned bits = word ? y1 : y0;
            float u = __uint_as_float((bits >> 9) | 0x3F800000u) - 1.0f;   // [0,1)
            const float TINY = 1.1754944e-38f;
            u = u * (1.0f - TINY) + TINY;
            if (u < TINY) u = TINY;
            const float gum = -__logf(-__logf(u));
            const float sc = logits[jj] + gum;
            if (sc > best) { best = sc; a = jj; }
        }
        float mx = logits[0];
        for (int jj = 1; jj < NACT; ++jj) mx = fmaxf(mx, logits[jj]);
        float s = 0.f;
        for (int jj = 0; jj < NACT; ++jj) s += __expf(logits[jj] - mx);
        const float lp = logits[a] - mx - __logf(s);

        out[b]         = (float)a;   // action
        out[BATCH + b] = lp;         // log_prob
    }
}

extern "C" void kernel_launch(void* const* d_in, const int* in_sizes, int n_in,
                              void* d_out, int out_size, void* d_ws, size_t ws_size,
                              hipStream_t stream) {
    (void)in_sizes; (void)n_in; (void)d_ws; (void)ws_size; (void)out_size;
    const float* x     = (const float*)d_in[0];  // (B, 12)
    const float* iscal = (const float*)d_in[1];  // (2, 12)
    const float* wts   = (const float*)d_in[2];  // (2, 24)
    const float* oscal = (const float*)d_in[3];  // (4,)
    float* out = (float*)d_out;                  // [actions(2048) | log_probs(2048)]
    pqc_reinforce_kernel<<<BATCH, 256, 0, stream>>>(x, iscal, wts, oscal, out);
}